// GLNonlin_28999619182851
// MI455X (gfx1250) — compile-verified
//
#include <hip/hip_runtime.h>

typedef float v2f __attribute__((ext_vector_type(2)));
typedef float v4f __attribute__((ext_vector_type(4)));
typedef float v8f __attribute__((ext_vector_type(8)));

#define BGDIM 128   // B*G = 8*16
#define ROWS  512   // M == N
#define DDIM  256   // D

// ---------------------------------------------------------------------------
// Kernel 1: column sums  s[bg*2+0][d] = sum_m u[bg,m,d]
//                        s[bg*2+1][d] = sum_n v[bg,n,d]
// grid.x = 2*BG, block = 1024 (d = tid&255 -> coalesced b32 streams,
// q = tid>>8 splits the 512 rows into 4 deterministic partials, LDS-combined).
// Default (RT) load policy on purpose: primes L2 with u/v for kernel 2.
// ---------------------------------------------------------------------------
__global__ __launch_bounds__(1024)
void colsum_kernel(const float* __restrict__ u, const float* __restrict__ v,
                   float* __restrict__ s) {
  const int bgm   = blockIdx.x;       // [0, 2*BG)
  const int which = bgm & 1;          // 0 -> u, 1 -> v
  const int bg    = bgm >> 1;
  const float* __restrict__ base =
      (which ? v : u) + (size_t)bg * ROWS * DDIM;

  const int t = threadIdx.x;
  const int d = t & (DDIM - 1);
  const int q = t >> 8;               // 0..3, 128 rows each
  const float* __restrict__ p = base + (size_t)(q * 128) * DDIM + d;

  float acc = 0.f;
#pragma unroll 8
  for (int m = 0; m < 128; ++m) acc += p[(size_t)m * DDIM];

  __shared__ float red[4][DDIM];
  red[q][d] = acc;
  __syncthreads();
  if (q == 0) {
    s[(size_t)bgm * DDIM + d] = red[0][d] + red[1][d] + red[2][d] + red[3][d];
  }
}

// ---------------------------------------------------------------------------
// Kernel 2: one wave per 16-row tile. row_sum = tile(16xD) . svec via
// V_WMMA_F32_16X16X4_F32 (A: float2/lane per ISA fp32 layout, B: svec
// broadcast across columns), then gate = (row_sum > 0) and coalesced gated
// b128 copy-out. Output stores are non-temporal: the result is never
// re-read, and keeping it out of L2 preserves the u/v working set (128 MB)
// inside the 192 MB L2 so the copy reads hit L2 instead of HBM.
// ---------------------------------------------------------------------------
__global__ __launch_bounds__(256)
void gate_kernel(const float* __restrict__ u, const float* __restrict__ v,
                 const float* __restrict__ s, float* __restrict__ out) {
  const int lane  = threadIdx.x & 31;
  const int wid   = (blockIdx.x << 3) + (threadIdx.x >> 5); // 0..8191
  const int bg    = wid >> 6;
  const int rem   = wid & 63;
  const int which = rem >> 5;    // 0 = u tile, 1 = v tile
  const int mtile = rem & 31;    // 16-row tile index

  const float* __restrict__ tile =
      (which ? v : u) + (size_t)bg * ROWS * DDIM + (size_t)mtile * 16 * DDIM;
  // gate u with s_v, gate v with s_u
  const float* __restrict__ svec = s + (size_t)(bg * 2 + (1 - which)) * DDIM;
  float* __restrict__ outp = out + (size_t)which * (size_t)BGDIM * ROWS * DDIM
                                 + (size_t)bg * ROWS * DDIM
                                 + (size_t)mtile * 16 * DDIM;

  // fp32 A-matrix 16x4 layout: M = lane&15 ; K = (lane>>4)*2 + vgpr
  const int arow = lane & 15;
  const int ak   = (lane >> 4) * 2;

  v8f acc = {0.f, 0.f, 0.f, 0.f, 0.f, 0.f, 0.f, 0.f};
#pragma unroll 4
  for (int d0 = 0; d0 < DDIM; d0 += 4) {
    v2f a = *(const v2f*)(tile + (size_t)arow * DDIM + d0 + ak); // K=ak,ak+1
    v2f b = *(const v2f*)(svec + d0 + ak);  // B[k][n] = svec[d0+k] for all n
    acc = __builtin_amdgcn_wmma_f32_16x16x4_f32(
        /*neg_a=*/false, a, /*neg_b=*/false, b,
        /*c_mod=*/(short)0, acc, /*reuse_a=*/false, /*reuse_b=*/false);
  }

  // D layout: VGPR r holds M=r (lanes 0-15) and M=8+r (lanes 16-31); every
  // column N is identical, so row m's sum sits in lane (m>>3)*16, vgpr m&7.
#pragma unroll
  for (int m = 0; m < 16; ++m) {
    float rs = __shfl(acc[m & 7], (m >> 3) * 16, 32);
    float g  = rs > 0.f ? 1.f : 0.f;      // relu(sign(x))
    const v4f p0 = *(const v4f*)(tile + (size_t)m * DDIM + lane * 8);
    const v4f p1 = *(const v4f*)(tile + (size_t)m * DDIM + lane * 8 + 4);
    v4f q0 = p0 * g;
    v4f q1 = p1 * g;
    __builtin_nontemporal_store(q0, (v4f*)(outp + (size_t)m * DDIM + lane * 8));
    __builtin_nontemporal_store(q1, (v4f*)(outp + (size_t)m * DDIM + lane * 8 + 4));
  }
}

extern "C" void kernel_launch(void* const* d_in, const int* in_sizes, int n_in,
                              void* d_out, int out_size, void* d_ws, size_t ws_size,
                              hipStream_t stream) {
  (void)in_sizes; (void)n_in; (void)out_size; (void)ws_size;
  const float* u = (const float*)d_in[0];
  const float* v = (const float*)d_in[1];
  float* out = (float*)d_out;
  float* s   = (float*)d_ws;   // 2*BG*D floats = 256 KB scratch

  colsum_kernel<<<2 * BGDIM, 1024, 0, stream>>>(u, v, s);

  const int tiles = BGDIM * 2 * (ROWS / 16);   // 8192 waves
  gate_kernel<<<tiles / 8, 256, 0, stream>>>(u, v, s, out);
}